// EdgeNodeAttention_65223373357283
// MI455X (gfx1250) — compile-verified
//
#include <hip/hip_runtime.h>
#include <hip/hip_bf16.h>

typedef __attribute__((ext_vector_type(2))) float v2f;
typedef __attribute__((ext_vector_type(8))) float v8f;

#define HID   64
#define NCLUS 64

__device__ __forceinline__ v8f wmma4(v2f a, v2f b, v8f c) {
  // D = A(16x4,f32) * B(4x16,f32) + C(16x16,f32)
  return __builtin_amdgcn_wmma_f32_16x16x4_f32(false, a, false, b, (short)0, c,
                                               false, false);
}

// float atomic max via monotonic int mapping (native global_atomic_max_i32 /
// global_atomic_min_u32; works because seg_max is initialized to -FLT_MAX)
__device__ __forceinline__ void atomicMaxF32(float* addr, float val) {
  if (val >= 0.0f)
    atomicMax((int*)addr, __float_as_int(val));
  else
    atomicMin((unsigned int*)addr, __float_as_uint(val));
}

__global__ void init_kernel(float* __restrict__ seg_max,
                            float* __restrict__ seg_sum, int n) {
  int i = blockIdx.x * blockDim.x + threadIdx.x;
  if (i < n) {
    seg_max[i] = -3.402823466e+38f;
    seg_sum[i] = 0.0f;
  }
}

// x_proj = x @ Wn + bn ; one wave per 16x16 output tile (4 waves/block over N-cols)
__global__ void node_proj_kernel(const float* __restrict__ x,
                                 const float* __restrict__ Wn,
                                 const float* __restrict__ bn,
                                 float* __restrict__ x_proj, int N) {
  const int wave = threadIdx.x >> 5;
  const int lane = threadIdx.x & 31;
  const int half = lane >> 4;
  const int l16  = lane & 15;
  const int m0 = blockIdx.x * 16;
  const int n0 = wave * 16;

  int m = m0 + l16;
  if (m >= N) m = N - 1;                 // clamp: keep EXEC all-ones for WMMA
  const float* xrow = x + (size_t)m * HID;

  v8f acc = {};
#pragma unroll
  for (int k0 = 0; k0 < HID; k0 += 4) {
    int ka = k0 + 2 * half;
    v2f a = *(const v2f*)(xrow + ka);                 // A[m][ka], A[m][ka+1]
    v2f b;
    b.x = Wn[ka * HID + n0 + l16];                    // B[ka][n]
    b.y = Wn[(ka + 1) * HID + n0 + l16];              // B[ka+1][n]
    acc = wmma4(a, b, acc);
  }
  float bias = bn[n0 + l16];
#pragma unroll
  for (int v = 0; v < 8; ++v) {
    int mo = m0 + v + 8 * half;                       // D: M = v + 8*half
    if (mo < N) x_proj[(size_t)mo * HID + n0 + l16] = acc[v] + bias;
  }
}

// Per 16-edge tile: att = (x_proj[col]+x_proj[row] + edge_attr@We + be) @ Wa + ba
// Writes att to out[], atomic-max into seg_max[col].
__global__ void edge_pass_a_kernel(const float* __restrict__ eattr,
                                   const int* __restrict__ eidx,
                                   const float* __restrict__ We,
                                   const float* __restrict__ be,
                                   const float* __restrict__ Wa,
                                   const float* __restrict__ ba,
                                   const float* __restrict__ x_proj,
                                   float* __restrict__ out,
                                   float* __restrict__ seg_max, int E) {
  __shared__ float hbuf[16 * 68];        // 16 edges x 64 hid, row stride 68 (bank-conflict pad)
  const int wave = threadIdx.x >> 5;
  const int lane = threadIdx.x & 31;
  const int half = lane >> 4;
  const int l16  = lane & 15;
  const int e0 = blockIdx.x * 16;
  const int n0 = wave * 16;              // this wave's 16-wide hid/cluster slice

  // ---- GEMM1: e_proj slice [16 edges x 16 hid] via WMMA f32 16x16x4
  int ea = e0 + l16;
  if (ea >= E) ea = E - 1;
  const float* arow = eattr + (size_t)ea * HID;
  v8f acc = {};
#pragma unroll
  for (int k0 = 0; k0 < HID; k0 += 4) {
    int ka = k0 + 2 * half;
    v2f a = *(const v2f*)(arow + ka);
    v2f b;
    b.x = We[ka * HID + n0 + l16];
    b.y = We[(ka + 1) * HID + n0 + l16];
    acc = wmma4(a, b, acc);
  }

  // ---- add bias + gathered node projections, stage h into LDS (same D layout,
  //      so everything here is thread-local before the barrier)
  float biasE = be[n0 + l16];
  int cols[8];
#pragma unroll
  for (int v = 0; v < 8; ++v) {
    int e = e0 + v + 8 * half;           // D layout: edge = v + 8*half
    if (e >= E) e = E - 1;
    int r = eidx[e];                     // edge_index[0][e]
    int c = eidx[E + e];                 // edge_index[1][e]
    cols[v] = c;
    float g = x_proj[(size_t)r * HID + n0 + l16] +
              x_proj[(size_t)c * HID + n0 + l16];
    hbuf[(v + 8 * half) * 68 + n0 + l16] = acc[v] + biasE + g;
  }
  __syncthreads();

  // ---- GEMM2: att slice = h @ Wa + ba (A from LDS via ds loads)
  v8f acc2 = {};
#pragma unroll
  for (int k0 = 0; k0 < HID; k0 += 4) {
    int ka = k0 + 2 * half;
    v2f a;
    a.x = hbuf[l16 * 68 + ka];           // A[edge=l16][ka]
    a.y = hbuf[l16 * 68 + ka + 1];
    v2f b;
    b.x = Wa[ka * NCLUS + n0 + l16];
    b.y = Wa[(ka + 1) * NCLUS + n0 + l16];
    acc2 = wmma4(a, b, acc2);
  }
  float biasA = ba[n0 + l16];
#pragma unroll
  for (int v = 0; v < 8; ++v) {
    int e = e0 + v + 8 * half;
    if (e >= E) e = E - 1;               // clamped dup writes same value: idempotent
    float val = acc2[v] + biasA;
    out[(size_t)e * NCLUS + n0 + l16] = val;
    atomicMaxF32(&seg_max[(size_t)cols[v] * NCLUS + n0 + l16], val);
  }
}

// out = exp(att - seg_max[col]); seg_sum[col] += out  (in-place on d_out)
__global__ void softmax_exp_kernel(float* __restrict__ out,
                                   const int* __restrict__ eidx,
                                   const float* __restrict__ seg_max,
                                   float* __restrict__ seg_sum, int E) {
  int idx = blockIdx.x * blockDim.x + threadIdx.x;
  if (idx >= E * NCLUS) return;
  int e = idx >> 6;
  int n = idx & 63;
  int c = eidx[E + e];
  float m = seg_max[(size_t)c * NCLUS + n];
  float v = expf(out[idx] - m);
  out[idx] = v;
  atomicAdd(&seg_sum[(size_t)c * NCLUS + n], v);
}

// out /= seg_sum[col] + 1e-16
__global__ void softmax_div_kernel(float* __restrict__ out,
                                   const int* __restrict__ eidx,
                                   const float* __restrict__ seg_sum, int E) {
  int idx = blockIdx.x * blockDim.x + threadIdx.x;
  if (idx >= E * NCLUS) return;
  int e = idx >> 6;
  int n = idx & 63;
  int c = eidx[E + e];
  out[idx] = out[idx] / (seg_sum[(size_t)c * NCLUS + n] + 1e-16f);
}

extern "C" void kernel_launch(void* const* d_in, const int* in_sizes, int n_in,
                              void* d_out, int out_size, void* d_ws, size_t ws_size,
                              hipStream_t stream) {
  const float* x     = (const float*)d_in[0];
  const int*   eidx  = (const int*)  d_in[1];   // [2,E] int32
  const float* eattr = (const float*)d_in[2];
  const float* Wn    = (const float*)d_in[3];
  const float* bn    = (const float*)d_in[4];
  const float* We    = (const float*)d_in[5];
  const float* be    = (const float*)d_in[6];
  const float* Wa    = (const float*)d_in[7];
  const float* ba    = (const float*)d_in[8];
  float* out = (float*)d_out;

  const int N = in_sizes[0] / HID;       // 50000
  const int E = in_sizes[2] / HID;       // 800000

  float* x_proj  = (float*)d_ws;                       // N*64 floats
  float* seg_max = x_proj  + (size_t)N * HID;          // N*64 floats
  float* seg_sum = seg_max + (size_t)N * NCLUS;        // N*64 floats

  int nc = N * NCLUS;
  init_kernel<<<(nc + 255) / 256, 256, 0, stream>>>(seg_max, seg_sum, nc);
  node_proj_kernel<<<(N + 15) / 16, 128, 0, stream>>>(x, Wn, bn, x_proj, N);
  edge_pass_a_kernel<<<(E + 15) / 16, 128, 0, stream>>>(eattr, eidx, We, be, Wa,
                                                        ba, x_proj, out, seg_max, E);
  int tot = E * NCLUS;
  softmax_exp_kernel<<<(tot + 255) / 256, 256, 0, stream>>>(out, eidx, seg_max,
                                                            seg_sum, E);
  softmax_div_kernel<<<(tot + 255) / 256, 256, 0, stream>>>(out, eidx, seg_sum, E);
}